// BaseLBSolver_40733469835707
// MI455X (gfx1250) — compile-verified
//
#include <hip/hip_runtime.h>

// ---------------------------------------------------------------------------
// LBM step for MI455X (gfx1250): pure HBM-bandwidth-bound streaming kernel.
// 212 B/site moved; design = wide NT b128 loads/stores + async G->LDS staging
// (ASYNCcnt) + scatter-streaming with power-of-two wrap. Wave32 throughout.
// ---------------------------------------------------------------------------

typedef float v4f __attribute__((ext_vector_type(4)));

#define QN     9
#define DIMY   2048
#define DIMX   2048
#define NPLANE ((size_t)DIMY * (size_t)DIMX)

__device__ __forceinline__ v4f ntload4(const float* p) {
  return __builtin_nontemporal_load((const v4f*)p);
}
__device__ __forceinline__ void ntstore4(float* p, v4f v) {
  __builtin_nontemporal_store(v, (v4f*)p);
}
__device__ __forceinline__ void ntstore1(float* p, float v) {
  __builtin_nontemporal_store(v, p);
}
__device__ __forceinline__ v4f vmaxs(v4f a, float b) {
  v4f r;
#pragma unroll
  for (int i = 0; i < 4; ++i) r[i] = fmaxf(a[i], b);
  return r;
}
__device__ __forceinline__ v4f vabs(v4f a) {
  v4f r;
#pragma unroll
  for (int i = 0; i < 4; ++i) r[i] = fabsf(a[i]);
  return r;
}
__device__ __forceinline__ v4f vdiv(v4f a, v4f b) {  // precise IEEE divide
  v4f r;
#pragma unroll
  for (int i = 0; i < 4; ++i) r[i] = a[i] / b[i];
  return r;
}
__device__ __forceinline__ v4f vrcp(v4f b) {
  v4f r;
#pragma unroll
  for (int i = 0; i < 4; ++i) r[i] = 1.0f / b[i];
  return r;
}

__global__ __launch_bounds__(256) void lbm_step_kernel(
    const float* __restrict__ F, const float* __restrict__ G,
    const float* __restrict__ Feq, float* __restrict__ out) {
  // D2Q9 velocity set
  constexpr int EXq[QN] = {1, 0, -1, 0, 1, -1, -1, 1, 0};
  constexpr int EYq[QN] = {0, 1, 0, -1, 1, 1, -1, -1, 0};

  __shared__ v4f gbuf[QN][256];  // 36 KB: block's G tile, staged by async DMA

  const int tx = threadIdx.x;                 // 0..63
  const int ty = threadIdx.y;                 // 0..3
  const int tid = ty * 64 + tx;               // 0..255
  const int x0 = (blockIdx.x * 64 + tx) * 4;  // base x (multiple of 4)
  const int y = blockIdx.y * 4 + ty;
  const size_t site = (size_t)y * DIMX + (size_t)x0;

  // ---- kick off async global->LDS copies of the G tile (ASYNCcnt path) ----
#pragma unroll
  for (int q = 0; q < QN; ++q) {
    const float* src = G + (size_t)q * NPLANE + site;
    unsigned lds = (unsigned)(size_t)(&gbuf[q][tid]);  // low 32b = LDS offset
    asm volatile("global_load_async_to_lds_b128 %0, %1, off" ::"v"(lds),
                 "v"(src)
                 : "memory");
  }

  // ---- F / Feq: read-once streaming -> non-temporal b128 loads ----
  v4f f[QN], fe[QN];
#pragma unroll
  for (int q = 0; q < QN; ++q) f[q] = ntload4(F + (size_t)q * NPLANE + site);
#pragma unroll
  for (int q = 0; q < QN; ++q) fe[q] = ntload4(Feq + (size_t)q * NPLANE + site);

  // ---- F-side moments + EPS (overlaps with async G DMA) ----
  v4f fsum = f[0];
#pragma unroll
  for (int q = 1; q < QN; ++q) fsum = fsum + f[q];
  v4f uxn = f[0] - f[2] + f[4] - f[5] - f[6] + f[7];  // EX weights (+UAX=0)
  v4f uyn = f[1] - f[3] + f[4] + f[5] - f[6] - f[7];  // EY weights (+UAY=0)

  v4f eps_sum = {0.f, 0.f, 0.f, 0.f};
#pragma unroll
  for (int q = 0; q < QN; ++q)
    eps_sum = eps_sum + vdiv(vabs(f[q] - fe[q]), fe[q] + 1e-10f);
  v4f EPS = eps_sum / 9.0f;

  v4f rho = vmaxs(fsum, 1e-6f);
  v4f inv_rho = vrcp(rho);
  v4f ux = uxn * inv_rho;
  v4f uy = uyn * inv_rho;

  // ---- consume staged G tile from LDS ----
  asm volatile("s_wait_asynccnt 0" ::: "memory");
  v4f gsum = gbuf[0][tid];
#pragma unroll
  for (int q = 1; q < QN; ++q) gsum = gsum + gbuf[q][tid];

  const float ICV = (float)(1.4 - 1.0);  // VUY - 1
  v4f E = gsum * 0.5f * inv_rho;
  v4f uu = ux * ux + uy * uy;
  v4f T = vmaxs(ICV * (E - uu * 0.5f), 1e-6f);

  v4f rhoH2 = 2.0f * rho * (E + T);
  v4f qx = rhoH2 * ux;
  v4f qy = rhoH2 * uy;

  v4f omT = 1.0f - T;
  v4f w04 = omT * T * 0.5f;   // planes 9..12
  v4f w47 = T * T * 0.25f;    // planes 13..16
  v4f w8 = omT * omT;         // plane 17

  // ---- adaptive relaxation (sequential-overwrite semantics collapse) ----
  v4f tau_DL = vdiv((v4f){0.01f, 0.01f, 0.01f, 0.01f}, rho * T) + 0.5f;
  v4f inv_tau_DL = vrcp(tau_DL);
  v4f alpha;
#pragma unroll
  for (int i = 0; i < 4; ++i)
    alpha[i] = (EPS[i] < 1.0f) ? 1.35f : inv_tau_DL[i];
  v4f tau = alpha * tau_DL;
  v4f tauT = 0.5f + (tau - 0.5f) / 0.71f;  // PR
  v4f omega = vrcp(tau);
  v4f omegaT = vrcp(tauT);

  // ---- BGK collision + scatter streaming (roll == scatter to y-ey, x+ex) --
#pragma unroll
  for (int q = 0; q < QN; ++q) {
    v4f fp = f[q] - omega * (f[q] - fe[q]);
    const int yo = (y - EYq[q]) & (DIMY - 1);
    float* dst = out + (size_t)q * NPLANE + (size_t)yo * DIMX;
    if (EXq[q] == 0) {
      ntstore4(dst + x0, fp);
    } else {
#pragma unroll
      for (int i = 0; i < 4; ++i)
        ntstore1(dst + (size_t)((x0 + i + EXq[q]) & (DIMX - 1)), fp[i]);
    }
  }

  // ---- w planes (9..17) ----
#pragma unroll
  for (int k = 0; k < 4; ++k)
    ntstore4(out + (size_t)(9 + k) * NPLANE + site, w04);
#pragma unroll
  for (int k = 0; k < 4; ++k)
    ntstore4(out + (size_t)(13 + k) * NPLANE + site, w47);
  ntstore4(out + (size_t)17 * NPLANE + site, w8);

  // ---- field planes (18..25): rho, ux, uy, E, T, qx, qy, omegaT ----
  ntstore4(out + (size_t)18 * NPLANE + site, rho);
  ntstore4(out + (size_t)19 * NPLANE + site, ux);
  ntstore4(out + (size_t)20 * NPLANE + site, uy);
  ntstore4(out + (size_t)21 * NPLANE + site, E);
  ntstore4(out + (size_t)22 * NPLANE + site, T);
  ntstore4(out + (size_t)23 * NPLANE + site, qx);
  ntstore4(out + (size_t)24 * NPLANE + site, qy);
  ntstore4(out + (size_t)25 * NPLANE + site, omegaT);
}

extern "C" void kernel_launch(void* const* d_in, const int* in_sizes, int n_in,
                              void* d_out, int out_size, void* d_ws,
                              size_t ws_size, hipStream_t stream) {
  (void)in_sizes; (void)n_in; (void)d_ws; (void)ws_size; (void)out_size;
  const float* F = (const float*)d_in[0];
  const float* G = (const float*)d_in[1];
  const float* Feq = (const float*)d_in[2];
  float* out = (float*)d_out;

  dim3 block(64, 4, 1);                       // 256 threads = 8 wave32
  dim3 grid(DIMX / (64 * 4), DIMY / 4, 1);    // 8 x 512 blocks
  lbm_step_kernel<<<grid, block, 0, stream>>>(F, G, Feq, out);
}